// GCN_76390288327439
// MI455X (gfx1250) — compile-verified
//
#include <hip/hip_runtime.h>
#include <math.h>

// ---------------------------------------------------------------------------
// GCN inference for MI455X (gfx1250, wave32).
// Dense GEMMs use V_WMMA_F32_16X16X4_F32 (fp32-exact matrix path).
// Graph aggregation (segment_sum) uses global float atomics (L2-resident).
// ---------------------------------------------------------------------------

typedef float v2f __attribute__((ext_vector_type(2)));
typedef float v8f __attribute__((ext_vector_type(8)));

#define WMMA_F32_K4(a, b, c)                                                   \
  __builtin_amdgcn_wmma_f32_16x16x4_f32(false, (a), false, (b), (short)0, (c), \
                                        false, false)

// ---------------------------------------------------------------- zero ws ---
__global__ void gcn_zero4(float4* __restrict__ p, size_t n4) {
  size_t i = (size_t)blockIdx.x * blockDim.x + threadIdx.x;
  size_t stride = (size_t)gridDim.x * blockDim.x;
  float4 z = make_float4(0.f, 0.f, 0.f, 0.f);
  for (; i < n4; i += stride) p[i] = z;
}

// ---------------------------------------------------------------- degrees ---
__global__ void gcn_degrees(const int* __restrict__ src,
                            const int* __restrict__ dst,
                            float* __restrict__ deg_src,
                            float* __restrict__ deg_dst, int E) {
  int e = blockIdx.x * blockDim.x + threadIdx.x;
  if (e < E) {
    atomicAdd(&deg_src[src[e]], 1.0f);
    atomicAdd(&deg_dst[dst[e]], 1.0f);
  }
}

// ------------------------------------------------------- deg -> 1/sqrt(d) ---
__global__ void gcn_norms(float* __restrict__ deg, int n2) {
  int i = blockIdx.x * blockDim.x + threadIdx.x;
  if (i < n2) {
    float d = deg[i];
    deg[i] = (d > 0.0f) ? rsqrtf(fmaxf(d, 1.0f)) : 0.0f;
  }
}

// -------------------------------------------------- GEMM1: (x*ns) @ W1 ------
// [N,128] @ [128,64] -> h1 [N,64].  Block = 128 thr = 4 waves.
// Wave w computes 16x16 tile: rows blockIdx*16.., cols 16*w..
template <bool GUARD>
__global__ void gcn_gemm1(const float* __restrict__ x,
                          const float* __restrict__ norm_src,
                          const float* __restrict__ W1, float* __restrict__ h1,
                          int n) {
  const int lane = threadIdx.x & 31;
  const int wave = threadIdx.x >> 5;
  const int half = lane >> 4;  // 0: K even-pair low, 1: high
  const int l16 = lane & 15;
  const int row0 = blockIdx.x * 16;
  const int col0 = wave * 16;
  const int rowA = GUARD ? min(row0 + l16, n - 1) : (row0 + l16);
  const float ns = norm_src[rowA];
  const float* xr = x + (size_t)rowA * 128;

  v8f acc = {};
#pragma unroll 4
  for (int k = 0; k < 128; k += 4) {
    const int kk = k + 2 * half;
    v2f a, b;
    a.x = xr[kk] * ns;
    a.y = xr[kk + 1] * ns;
    b.x = W1[(size_t)kk * 64 + col0 + l16];
    b.y = W1[(size_t)(kk + 1) * 64 + col0 + l16];
    acc = WMMA_F32_K4(a, b, acc);
  }
#pragma unroll
  for (int r = 0; r < 8; ++r) {
    int row = row0 + r + 8 * half;
    if (!GUARD || row < n) h1[(size_t)row * 64 + col0 + l16] = acc[r];
  }
}

// ----------------------------------------------- scatter-add over edges -----
// agg[dst[e]][f] += h[src[e]][f]
template <int F>
__global__ void gcn_scatter(const int* __restrict__ src,
                            const int* __restrict__ dst,
                            const float* __restrict__ h,
                            float* __restrict__ agg, int E) {
  unsigned i = blockIdx.x * blockDim.x + threadIdx.x;
  unsigned total = (unsigned)E * F;
  if (i < total) {
    unsigned e = i / F;
    unsigned f = i - e * F;
    atomicAdd(&agg[(size_t)dst[e] * F + f], h[(size_t)src[e] * F + f]);
  }
}

// --------- GEMM2: (ns * relu(agg1*nd + b1)) @ W2  -> h2 [N,32] --------------
// Block = 128 thr = 4 waves covering 32 rows x 32 cols.
template <bool GUARD>
__global__ void gcn_gemm2(const float* __restrict__ agg1,
                          const float* __restrict__ norm_src,
                          const float* __restrict__ norm_dst,
                          const float* __restrict__ b1,
                          const float* __restrict__ W2, float* __restrict__ h2,
                          int n) {
  const int lane = threadIdx.x & 31;
  const int wave = threadIdx.x >> 5;
  const int half = lane >> 4;
  const int l16 = lane & 15;
  const int row0 = blockIdx.x * 32 + (wave >> 1) * 16;
  const int col0 = (wave & 1) * 16;
  const int rowA = GUARD ? min(row0 + l16, n - 1) : (row0 + l16);
  const float ns = norm_src[rowA];
  const float nd = norm_dst[rowA];
  const float* ar = agg1 + (size_t)rowA * 64;

  v8f acc = {};
#pragma unroll 4
  for (int k = 0; k < 64; k += 4) {
    const int kk = k + 2 * half;
    v2f a, b;
    a.x = fmaxf(ar[kk] * nd + b1[kk], 0.0f) * ns;
    a.y = fmaxf(ar[kk + 1] * nd + b1[kk + 1], 0.0f) * ns;
    b.x = W2[(size_t)kk * 32 + col0 + l16];
    b.y = W2[(size_t)(kk + 1) * 32 + col0 + l16];
    acc = WMMA_F32_K4(a, b, acc);
  }
#pragma unroll
  for (int r = 0; r < 8; ++r) {
    int row = row0 + r + 8 * half;
    if (!GUARD || row < n) h2[(size_t)row * 32 + col0 + l16] = acc[r];
  }
}

// ---------- final: out[i] = sum_f (agg2[i][f]*nd[i] + b2[f]) * Wl[f] + bl ---
// One wave32 per node, lane == feature, shuffle reduction.
__global__ void gcn_final(const float* __restrict__ agg2,
                          const float* __restrict__ norm_dst,
                          const float* __restrict__ b2,
                          const float* __restrict__ Wl,
                          const float* __restrict__ bl, float* __restrict__ out,
                          int n) {
  const int gtid = blockIdx.x * blockDim.x + threadIdx.x;
  const int node = gtid >> 5;
  const int lane = gtid & 31;
  if (node >= n) return;
  float v = agg2[(size_t)node * 32 + lane] * norm_dst[node] + b2[lane];
  float p = v * Wl[lane];
#pragma unroll
  for (int off = 16; off > 0; off >>= 1) p += __shfl_xor(p, off, 32);
  if (lane == 0) out[node] = p + bl[0];
}

// ---------------------------------------------------------------------------
extern "C" void kernel_launch(void* const* d_in, const int* in_sizes, int n_in,
                              void* d_out, int out_size, void* d_ws,
                              size_t ws_size, hipStream_t stream) {
  const float* x = (const float*)d_in[0];
  const int* ei = (const int*)d_in[1];
  const float* W1 = (const float*)d_in[2];
  const float* b1 = (const float*)d_in[3];
  const float* W2 = (const float*)d_in[4];
  const float* b2 = (const float*)d_in[5];
  const float* Wl = (const float*)d_in[6];
  const float* bl = (const float*)d_in[7];
  float* out = (float*)d_out;

  const int n = in_sizes[0] / 128;  // 100000
  const int E = in_sizes[1] / 2;    // 1600000
  const int* src = ei;
  const int* dst = ei + E;

  // workspace layout (floats): deg/norm_src n | deg/norm_dst n |
  //   h1 64n | agg1 64n | h2 32n | agg2 32n   -> 194n floats (~77.6 MB)
  float* ws = (float*)d_ws;
  float* norm_src = ws;
  float* norm_dst = ws + n;
  float* h1 = ws + 2 * (size_t)n;
  float* agg1 = h1 + 64 * (size_t)n;
  float* h2 = agg1 + 64 * (size_t)n;
  float* agg2 = h2 + 32 * (size_t)n;
  const size_t total_ws = 194 * (size_t)n;  // divisible by 2; pad to /4

  // 1) zero degrees + accumulation buffers (zero whole region; cheap)
  gcn_zero4<<<4096, 256, 0, stream>>>((float4*)ws, (total_ws + 3) / 4);

  // 2) degrees via float atomics
  gcn_degrees<<<(E + 255) / 256, 256, 0, stream>>>(src, dst, norm_src, norm_dst,
                                                   E);

  // 3) deg -> 1/sqrt(deg) in place (both arrays, contiguous)
  gcn_norms<<<(2 * n + 255) / 256, 256, 0, stream>>>(ws, 2 * n);

  // 4) h1 = (x * norm_src) @ W1   (WMMA fp32)
  if (n % 16 == 0)
    gcn_gemm1<false><<<n / 16, 128, 0, stream>>>(x, norm_src, W1, h1, n);
  else
    gcn_gemm1<true><<<(n + 15) / 16, 128, 0, stream>>>(x, norm_src, W1, h1, n);

  // 5) agg1[dst] += h1[src]  (64 features)
  {
    unsigned work = (unsigned)E * 64u;
    gcn_scatter<64><<<(work + 255) / 256, 256, 0, stream>>>(src, dst, h1, agg1,
                                                            E);
  }

  // 6) h2 = (norm_src * relu(agg1*norm_dst + b1)) @ W2   (WMMA fp32, fused)
  if (n % 32 == 0)
    gcn_gemm2<false><<<n / 32, 128, 0, stream>>>(agg1, norm_src, norm_dst, b1,
                                                 W2, h2, n);
  else
    gcn_gemm2<true><<<(n + 31) / 32, 128, 0, stream>>>(agg1, norm_src, norm_dst,
                                                       b1, W2, h2, n);

  // 7) agg2[dst] += h2[src]  (32 features)
  {
    unsigned work = (unsigned)E * 32u;
    gcn_scatter<32><<<(work + 255) / 256, 256, 0, stream>>>(src, dst, h2, agg2,
                                                            E);
  }

  // 8) out = (agg2*norm_dst + b2) @ Wl + bl
  gcn_final<<<((n * 32) + 255) / 256, 256, 0, stream>>>(agg2, norm_dst, b2, Wl,
                                                        bl, out, n);
}